// Attention_1073741824210
// MI455X (gfx1250) — compile-verified
//
#include <hip/hip_runtime.h>
#include <math.h>

typedef __attribute__((ext_vector_type(2))) float v2f;
typedef __attribute__((ext_vector_type(8))) float v8f;

#define B_SZ   32
#define N_SZ   1024
#define D_SZ   64
#define SCALE  0.125f   // 1/sqrt(64)

#define WMMA_F32(A, B, C) \
    __builtin_amdgcn_wmma_f32_16x16x4_f32(false, (A), false, (B), (short)0, (C), false, false)

// ---- One-shot KR = K + R (8 MB into d_ws; pure bandwidth, ~1 us) ----
__global__ __launch_bounds__(256)
void add_kr_kernel(const float4* __restrict__ K, const float4* __restrict__ R,
                   float4* __restrict__ KR)
{
    const int i = blockIdx.x * 256 + threadIdx.x;
    const float4 k = K[i];
    const float4 r = R[i];
    float4 o;
    o.x = k.x + r.x; o.y = k.y + r.y; o.z = k.z + r.z; o.w = k.w + r.w;
    KR[i] = o;
}

// One wave32 per 16 query rows; 64-key chunks. Flash-style online softmax with
// WMMA-computed row sums (P @ ones) and chunk-amortized max butterflies.
// PRE=true: Kp already holds K+R (from d_ws). PRE=false: fuse the add in-loop.
template<bool PRE>
__global__ __launch_bounds__(32)
void attn_fused_f32_wmma(const float* __restrict__ Q, const float* __restrict__ Kp,
                         const float* __restrict__ V, const float* __restrict__ Rp,
                         float* __restrict__ outO, float* __restrict__ outS)
{
    __shared__ __align__(16) float pLds[16 * 64];   // 4 KB P chunk

    const int lane = threadIdx.x & 31;
    const int hi   = lane >> 4;    // 16-lane half
    const int ln   = lane & 15;
    const int b    = blockIdx.x >> 6;            // / (N/16)
    const int row0 = (blockIdx.x & 63) << 4;     // 16-row block

    const float2* Q2 = (const float2*)Q;
    const float2* K2 = (const float2*)Kp;
    const float2* R2 = (const float2*)Rp;

    // ---- Resident Q tile (16x64) in A-layout, pre-scaled by 1/sqrt(D) ----
    const size_t qbase2 = (((size_t)b * N_SZ + row0 + ln) * D_SZ) >> 1;
    v2f qa[16];
#pragma unroll
    for (int s = 0; s < 16; ++s) {
        float2 t = Q2[qbase2 + 2 * s + hi];
        v2f a = {t.x * SCALE, t.y * SCALE};
        qa[s] = a;
    }

    // ---- Online softmax state: this lane owns rows 8*hi + r ----
    float mrow[8];
#pragma unroll
    for (int r = 0; r < 8; ++r) mrow[r] = -INFINITY;

    v8f acc0 = {}, acc1 = {}, acc2 = {}, acc3 = {};
    v8f accL = {};                       // row-sum accumulator (P @ ones)
    const v2f vones = {1.0f, 1.0f};

    const size_t kvbase = (size_t)b * N_SZ * D_SZ;

    for (int mc = 0; mc < N_SZ / 64; ++mc) {
        const int m0 = mc << 6;

        // ---- 4 score tiles: S = (Q*scale) (K+R)^T ----
        v8f c[4];
#pragma unroll
        for (int t = 0; t < 4; ++t) {
            v8f ct = {};
            const size_t kb2 = (kvbase + (size_t)(m0 + 16 * t + ln) * D_SZ) >> 1;
#pragma unroll
            for (int s = 0; s < 16; ++s) {
                float2 kk = K2[kb2 + 2 * s + hi];
                v2f bb;
                if constexpr (PRE) {
                    bb = (v2f){kk.x, kk.y};
                } else {
                    float2 rr = R2[kb2 + 2 * s + hi];
                    bb = (v2f){kk.x + rr.x, kk.y + rr.y};
                }
                ct = WMMA_F32(qa[s], bb, ct);
            }
            c[t] = ct;
            // stream out attention_score tile (write-once -> non-temporal)
            const size_t srow = ((size_t)b * N_SZ + row0 + 8 * hi) * N_SZ
                              + m0 + 16 * t + ln;
#pragma unroll
            for (int r = 0; r < 8; ++r)
                __builtin_nontemporal_store(ct[r], &outS[srow + (size_t)r * N_SZ]);
        }

        // ---- Chunk-wide row max (one butterfly per 64 keys) ----
        float alpha[8], mn[8];
#pragma unroll
        for (int r = 0; r < 8; ++r) {
            float t = fmaxf(fmaxf(c[0][r], c[1][r]), fmaxf(c[2][r], c[3][r]));
            t = fmaxf(t, __shfl_xor(t, 1, 16));
            t = fmaxf(t, __shfl_xor(t, 2, 16));
            t = fmaxf(t, __shfl_xor(t, 4, 16));
            t = fmaxf(t, __shfl_xor(t, 8, 16));
            const float m2 = fmaxf(mrow[r], t);
            alpha[r] = __expf(mrow[r] - m2);
            mn[r] = m2;
            mrow[r] = m2;
        }

        // ---- Exponentiate in place ----
#pragma unroll
        for (int t = 0; t < 4; ++t)
#pragma unroll
            for (int r = 0; r < 8; ++r)
                c[t][r] = __expf(c[t][r] - mn[r]);

        // ---- Rescale running accumulators (incl. row-sum column) ----
#pragma unroll
        for (int r = 0; r < 8; ++r) {
            acc0[r] *= alpha[r]; acc1[r] *= alpha[r];
            acc2[r] *= alpha[r]; acc3[r] *= alpha[r];
            accL[r] *= alpha[r];
        }

        // ---- P chunk: C-layout -> LDS -> A-layout ----
#pragma unroll
        for (int t = 0; t < 4; ++t)
#pragma unroll
            for (int r = 0; r < 8; ++r)
                pLds[(8 * hi + r) * 64 + 16 * t + ln] = c[t][r];
        __syncthreads();

        // ---- O += P V (and rowsum += P @ 1): 16 k-steps x (4 d-tiles + 1) ----
        const float2* pl2 = (const float2*)pLds;
        const float*  Vb  = V + kvbase + (size_t)m0 * D_SZ;
#pragma unroll
        for (int s = 0; s < 16; ++s) {
            float2 tp = pl2[ln * 32 + 2 * s + hi];   // element ln*64 + 4s + 2hi
            v2f pa = {tp.x, tp.y};
            const float* vr = Vb + (4 * s + 2 * hi) * D_SZ + ln;
            v2f vb;
            vb = (v2f){vr[0],  vr[D_SZ + 0]};  acc0 = WMMA_F32(pa, vb, acc0);
            vb = (v2f){vr[16], vr[D_SZ + 16]}; acc1 = WMMA_F32(pa, vb, acc1);
            vb = (v2f){vr[32], vr[D_SZ + 32]}; acc2 = WMMA_F32(pa, vb, acc2);
            vb = (v2f){vr[48], vr[D_SZ + 48]}; acc3 = WMMA_F32(pa, vb, acc3);
            accL = WMMA_F32(pa, vones, accL);
        }
        __syncthreads();   // protect pLds before next chunk's writes
    }

    // ---- Epilogue: normalize by WMMA-computed row sums, store O (NT) ----
    float rl[8];
#pragma unroll
    for (int r = 0; r < 8; ++r) rl[r] = 1.0f / accL[r];
    const size_t obase = ((size_t)b * N_SZ + row0 + 8 * hi) * D_SZ + ln;
#pragma unroll
    for (int r = 0; r < 8; ++r) {
        __builtin_nontemporal_store(acc0[r] * rl[r], &outO[obase + (size_t)r * D_SZ + 0 ]);
        __builtin_nontemporal_store(acc1[r] * rl[r], &outO[obase + (size_t)r * D_SZ + 16]);
        __builtin_nontemporal_store(acc2[r] * rl[r], &outO[obase + (size_t)r * D_SZ + 32]);
        __builtin_nontemporal_store(acc3[r] * rl[r], &outO[obase + (size_t)r * D_SZ + 48]);
    }
}

extern "C" void kernel_launch(void* const* d_in, const int* in_sizes, int n_in,
                              void* d_out, int out_size, void* d_ws, size_t ws_size,
                              hipStream_t stream) {
    const float* Q = (const float*)d_in[0];
    const float* K = (const float*)d_in[1];
    const float* V = (const float*)d_in[2];
    const float* R = (const float*)d_in[3];
    float* outO = (float*)d_out;                                   // [32,1024,64]
    float* outS = outO + (size_t)B_SZ * N_SZ * D_SZ;               // [32,1024,1024]

    const size_t kr_elems = (size_t)B_SZ * N_SZ * D_SZ;            // 2M floats
    const size_t kr_bytes = kr_elems * sizeof(float);              // 8 MB
    dim3 grid(B_SZ * (N_SZ / 16));   // 2048 waves, one per 16 query rows

    if (ws_size >= kr_bytes) {
        // Stage 1: KR = K + R once (stream-ordered before attention).
        add_kr_kernel<<<(unsigned)(kr_elems / 4 / 256), 256, 0, stream>>>(
            (const float4*)K, (const float4*)R, (float4*)d_ws);
        attn_fused_f32_wmma<true><<<grid, 32, 0, stream>>>(
            Q, (const float*)d_ws, V, R, outO, outS);
    } else {
        // Fallback: fuse K+R in the score loop.
        attn_fused_f32_wmma<false><<<grid, 32, 0, stream>>>(
            Q, K, V, R, outO, outS);
    }
}